// Model_1640677507274
// MI455X (gfx1250) — compile-verified
//
#include <hip/hip_runtime.h>
#include <math.h>

// GAT layer on gfx1250 (MI455X): fp32 WMMA (V_WMMA_F32_16X16X4_F32) for both
// GEMMs, fused online-softmax attention, wave32.
//
// v2 changes:
//  - h_prime stored pair-interleaved (B,H,N/2,F,2) so each lane's two
//    B-operand K-values are one global_load_b64.
//  - attention probabilities computed ONCE per workgroup into LDS (pitch-17
//    pair layout, bank-conflict-free; A-operand = one 8B LDS read/lane),
//    removing the 4x duplicated exp/leaky/mask VALU work.
//  - s_i/s_j fused into the projection GEMM (shfl-xor reduce of C fragments).

typedef __attribute__((ext_vector_type(2))) float v2f;
typedef __attribute__((ext_vector_type(8))) float v8f;

#define GAT_B 8
#define GAT_N 1024
#define GAT_F 128
#define GAT_H 8
#define NEG_SLOPE 0.2f
#define VERY_NEG (-1.0e30f)
#define PPITCH 17          // padded i-pitch for the LDS probability tile
#define JCHUNK 512         // j-chunk held in LDS (256 pairs)

static __device__ __forceinline__ v8f wmma_f32(v2f a, v2f b, v8f c) {
  return __builtin_amdgcn_wmma_f32_16x16x4_f32(
      /*neg_a=*/false, a, /*neg_b=*/false, b,
      /*c_mod=*/(short)0, c, /*reuse_a=*/false, /*reuse_b=*/false);
}

// ---------------------------------------------------------------------------
// Kernel 1: h_prime[b,h,n,f] = sum_k h[b,n,k] * W[h,k,f]   (+ fused scores)
// grid = B*(N/16)*H workgroups, 128 threads (4 waves), each wave 16x32.
// Output layout: pair-interleaved (B,H,N/2,F,2).
// Also emits s1[b,h,n] = hp_row . a1[h], s2 = hp_row . a2[h].
// ---------------------------------------------------------------------------
__global__ __launch_bounds__(128) void gat_gemm_kernel(
    const float* __restrict__ h,    // (B,N,128)
    const float* __restrict__ W,    // (H,128,128)
    const float* __restrict__ avec, // (H,256,1)
    float* __restrict__ hp,         // (B,H,N/2,128,2)
    float* __restrict__ s1,         // (B,H,N)
    float* __restrict__ s2)
{
  const int hd    = blockIdx.x & 7;
  const int ntile = (blockIdx.x >> 3) & 63;
  const int b     = blockIdx.x >> 9;
  const int lane  = threadIdx.x & 31;
  const int wave  = threadIdx.x >> 5;

  const int n0  = ntile * 16;
  const int M   = lane & 15;         // A-matrix row for this lane
  const int kb  = (lane >> 4) * 2;   // K pair base: 0 (lanes 0-15) / 2 (16-31)
  const int f0  = wave * 32;
  const int col = f0 + (lane & 15);

  __shared__ float red1[4][16];
  __shared__ float red2[4][16];

  const float* __restrict__ hrow = h + ((size_t)b * GAT_N + n0 + M) * GAT_F;
  const float* __restrict__ wmat = W + (size_t)hd * GAT_F * GAT_F;

  v8f c0 = {}; v8f c1 = {};
  for (int k = 0; k < GAT_F; k += 4) {
    v2f a;
    a.x = hrow[k + kb];
    a.y = hrow[k + kb + 1];
    v2f b0, b1;
    b0.x = wmat[(size_t)(k + kb)     * GAT_F + col];
    b0.y = wmat[(size_t)(k + kb + 1) * GAT_F + col];
    b1.x = wmat[(size_t)(k + kb)     * GAT_F + col + 16];
    b1.y = wmat[(size_t)(k + kb + 1) * GAT_F + col + 16];
    c0 = wmma_f32(a, b0, c0);
    c1 = wmma_f32(a, b1, c1);
  }

  // ---- store pair-interleaved panel ----
  // C/D layout: VGPR r holds row M=r (lanes 0-15) and M=r+8 (lanes 16-31).
  const size_t bh = (size_t)b * GAT_H + hd;
  float* __restrict__ panel = hp + bh * (GAT_N / 2) * GAT_F * 2;
  const int rb = (lane >> 4) * 8;
  for (int r = 0; r < 8; ++r) {
    const int grow = n0 + rb + r;
    const size_t pbase = ((size_t)(grow >> 1) * GAT_F);
    panel[(pbase + col)      * 2 + (grow & 1)] = c0[r];
    panel[(pbase + col + 16) * 2 + (grow & 1)] = c1[r];
  }

  // ---- fused attention scores ----
  const float* __restrict__ a1 = avec + (size_t)hd * 256;
  const float* __restrict__ a2 = a1 + 128;
  const float a10 = a1[col], a11 = a1[col + 16];
  const float a20 = a2[col], a21 = a2[col + 16];
  for (int r = 0; r < 8; ++r) {
    float v1 = c0[r] * a10 + c1[r] * a11;
    float v2 = c0[r] * a20 + c1[r] * a21;
    // reduce over the 16 columns held by each 16-lane half
    for (int off = 8; off > 0; off >>= 1) {
      v1 += __shfl_xor(v1, off, 32);
      v2 += __shfl_xor(v2, off, 32);
    }
    if ((lane & 15) == 0) {            // lane 0 -> row r, lane 16 -> row r+8
      red1[wave][rb + r] = v1;
      red2[wave][rb + r] = v2;
    }
  }
  __syncthreads();
  if (threadIdx.x < 16) {
    const int t = threadIdx.x;
    float t1 = red1[0][t] + red1[1][t] + red1[2][t] + red1[3][t];
    float t2 = red2[0][t] + red2[1][t] + red2[2][t] + red2[3][t];
    s1[bh * GAT_N + n0 + t] = t1;
    s2[bh * GAT_N + n0 + t] = t2;
  }
}

// ---------------------------------------------------------------------------
// Kernel 2: fused masked softmax + (attention @ h_prime).
// One workgroup per (b, head, i-tile of 16 rows); 4 waves.
// Pass 1: online softmax stats (m_i, l_i) over all j (waves split j-range).
// Per 512-j chunk:
//   2a: whole WG materializes the 16x512 probability tile ONCE into LDS
//       (pitch-17 pair layout: P[(jj/2)*17 + i][2], bank-conflict-free).
//   2b: each wave owns 2 f-tiles; A-operand = 8B LDS read, B-operand =
//       two global_load_b64 from the pair-interleaved h_prime panel.
// ---------------------------------------------------------------------------
__global__ __launch_bounds__(128) void gat_attn_kernel(
    const float* __restrict__ hp,    // (B,H,N/2,128,2)
    const float* __restrict__ s1,    // (B,H,N)
    const float* __restrict__ s2,    // (B,H,N)
    const float* __restrict__ bias,  // (N,N)
    const int*   __restrict__ adj,   // (N,N)
    float* __restrict__ hout)        // (B,H,N,128)
{
  const int itile = blockIdx.x & 63;
  const int hd    = (blockIdx.x >> 6) & 7;
  const int b     = blockIdx.x >> 9;
  const int i0    = itile * 16;
  const int lane  = threadIdx.x & 31;
  const int wave  = threadIdx.x >> 5;

  __shared__ float P[(JCHUNK / 2) * PPITCH * 2];   // 8704 floats = 34.8 KB
  __shared__ float sj[GAT_N];
  __shared__ float si[16];
  __shared__ float red_m[4 * 16];
  __shared__ float red_l[4 * 16];
  __shared__ float mi[16];
  __shared__ float inv_li[16];

  const size_t bh = (size_t)b * GAT_H + hd;
  const float* __restrict__ s2row = s2 + bh * GAT_N;
  for (int t = threadIdx.x; t < GAT_N; t += 128) sj[t] = s2row[t];
  if (threadIdx.x < 16)
    si[threadIdx.x] = s1[bh * GAT_N + i0 + threadIdx.x];
  __syncthreads();

  // ---- pass 1: per-row max & exp-sum ----
  const int jbase = wave * 256;
  for (int r = 0; r < 16; ++r) {
    const float sir = si[r];
    const float* __restrict__ brow = bias + (size_t)(i0 + r) * GAT_N + jbase;
    const int*   __restrict__ arow = adj  + (size_t)(i0 + r) * GAT_N + jbase;
    float m = VERY_NEG, l = 0.0f;
    for (int t = lane; t < 256; t += 32) {
      if (arow[t] != 0) {
        float x = sir + sj[jbase + t];
        float e = (x > 0.0f ? x : NEG_SLOPE * x) + brow[t];
        float mn = fmaxf(m, e);
        l = l * __expf(m - mn) + __expf(e - mn);
        m = mn;
      }
    }
    for (int off = 16; off > 0; off >>= 1) {
      float mo = __shfl_xor(m, off, 32);
      float lo = __shfl_xor(l, off, 32);
      float mn = fmaxf(m, mo);
      l = l * __expf(m - mn) + lo * __expf(mo - mn);
      m = mn;
    }
    if (lane == 0) { red_m[wave * 16 + r] = m; red_l[wave * 16 + r] = l; }
  }
  __syncthreads();
  if (threadIdx.x < 16) {
    float m = red_m[threadIdx.x];
    float l = red_l[threadIdx.x];
    for (int w = 1; w < 4; ++w) {
      float mo = red_m[w * 16 + threadIdx.x];
      float lo = red_l[w * 16 + threadIdx.x];
      float mn = fmaxf(m, mo);
      l = l * __expf(m - mn) + lo * __expf(mo - mn);
      m = mn;
    }
    mi[threadIdx.x]     = m;
    inv_li[threadIdx.x] = (l > 0.0f) ? (1.0f / l) : 0.0f;
  }
  __syncthreads();

  // ---- pass 2: chunked probability tile + WMMA ----
  const float2* __restrict__ hp2 =
      (const float2*)hp + bh * (GAT_N / 2) * GAT_F;   // (N/2, F) of float2
  const int i   = lane & 15;
  const int kb  = (lane >> 4) * 2;   // A-operand K pair base (ISA layout)
  const int f0  = wave * 32;
  const int col = f0 + (lane & 15);

  v8f c0 = {}; v8f c1 = {};
  for (int c = 0; c < GAT_N / JCHUNK; ++c) {
    const int jc = c * JCHUNK;

    // 2a: materialize probabilities for this chunk once per workgroup
    for (int idx = threadIdx.x; idx < 16 * JCHUNK; idx += 128) {
      const int ii = idx >> 9;          // 0..15
      const int jj = idx & (JCHUNK - 1);
      const int j  = jc + jj;
      float p = 0.0f;
      if (adj[(size_t)(i0 + ii) * GAT_N + j] != 0) {
        float x = si[ii] + sj[j];
        float e = (x > 0.0f ? x : NEG_SLOPE * x) +
                  bias[(size_t)(i0 + ii) * GAT_N + j];
        p = __expf(e - mi[ii]) * inv_li[ii];
      }
      P[((jj >> 1) * PPITCH + ii) * 2 + (jj & 1)] = p;
    }
    __syncthreads();

    // 2b: 16x16x4 WMMAs over the chunk
    for (int jb = 0; jb < JCHUNK; jb += 4) {
      const int jj0 = jb + kb;          // even
      const float2 pv = *(const float2*)&P[((jj0 >> 1) * PPITCH + i) * 2];
      v2f pa; pa.x = pv.x; pa.y = pv.y;

      const size_t prow = (size_t)((jc + jj0) >> 1) * GAT_F;
      const float2 b0v = hp2[prow + col];
      const float2 b1v = hp2[prow + col + 16];
      v2f b0; b0.x = b0v.x; b0.y = b0v.y;
      v2f b1; b1.x = b1v.x; b1.y = b1v.y;

      c0 = wmma_f32(pa, b0, c0);
      c1 = wmma_f32(pa, b1, c1);
    }
    __syncthreads();   // P reused next chunk
  }

  // store (row-major hout): VGPR r holds rows r / r+8 per 16-lane half
  float* __restrict__ orow = hout + (bh * GAT_N + i0) * GAT_F;
  const int rb = (lane >> 4) * 8;
  for (int r = 0; r < 8; ++r) {
    orow[(size_t)(rb + r) * GAT_F + col]      = c0[r];
    orow[(size_t)(rb + r) * GAT_F + col + 16] = c1[r];
  }
}

// ---------------------------------------------------------------------------
// Kernel 3: out[b,n,f] = mean over heads of hout[b,h,n,f]
// ---------------------------------------------------------------------------
__global__ __launch_bounds__(256) void gat_mean_kernel(
    const float* __restrict__ hout,  // (B,H,N,128)
    float* __restrict__ out)         // (B,N,128)
{
  const int idx = blockIdx.x * 256 + threadIdx.x;  // B*N*F = 1M
  const int f = idx & 127;
  const int n = (idx >> 7) & 1023;
  const int b = idx >> 17;
  float s = 0.0f;
  for (int hd = 0; hd < GAT_H; ++hd)
    s += hout[(((size_t)b * GAT_H + hd) * GAT_N + n) * GAT_F + f];
  out[idx] = s * 0.125f;
}

// ---------------------------------------------------------------------------
extern "C" void kernel_launch(void* const* d_in, const int* in_sizes, int n_in,
                              void* d_out, int out_size, void* d_ws, size_t ws_size,
                              hipStream_t stream) {
  const float* h    = (const float*)d_in[0];  // (B,N,F_IN)
  const int*   adj  = (const int*)  d_in[1];  // (N,N)
  const float* bias = (const float*)d_in[2];  // (1,1,N,N)
  const float* W    = (const float*)d_in[3];  // (H,F_IN,F_OUT)
  const float* a    = (const float*)d_in[4];  // (H,2F,1)
  float* out = (float*)d_out;

  char* ws = (char*)d_ws;
  const size_t hpBytes = (size_t)GAT_B * GAT_H * GAT_N * GAT_F * sizeof(float); // 32MB
  const size_t sBytes  = (size_t)GAT_B * GAT_H * GAT_N * sizeof(float);         // 256KB
  float* hp   = (float*)ws;
  float* s1v  = (float*)(ws + hpBytes);
  float* s2v  = (float*)(ws + hpBytes + sBytes);
  float* hout = (float*)(ws + hpBytes + 2 * sBytes);

  // 1) h_prime = h @ W (per head) + fused s_i/s_j
  gat_gemm_kernel<<<dim3(GAT_B * (GAT_N / 16) * GAT_H), dim3(128), 0, stream>>>(
      h, W, a, hp, s1v, s2v);

  // 2) fused masked softmax + attention @ h_prime
  gat_attn_kernel<<<dim3(GAT_B * GAT_H * (GAT_N / 16)), dim3(128), 0, stream>>>(
      hp, s1v, s2v, bias, adj, hout);

  // 3) mean over heads
  gat_mean_kernel<<<dim3(GAT_B * GAT_N * GAT_F / 256), dim3(256), 0, stream>>>(
      hout, out);
}